// GNNModel_14173392077522
// MI455X (gfx1250) — compile-verified
//
#include <hip/hip_runtime.h>
#include <math.h>

#define P_   100
#define H_   4
#define C_   128
#define E_   128
#define HC_  512
#define OUT_ 128
#define S_   65536
#define PPAD 104   // hp zero-padded to 104 rows so the K-loop needs no masking
#define MT   64    // node rows per block (4 WMMA m-tiles)

typedef __attribute__((ext_vector_type(2))) float v2f;
typedef __attribute__((ext_vector_type(8))) float v8f;

__device__ __forceinline__ v8f wmma_f32(v2f a, v2f b, v8f c) {
  // V_WMMA_F32_16X16X4_F32: D = A(16x4) x B(4x16) + C(16x16)
  return __builtin_amdgcn_wmma_f32_16x16x4_f32(
      /*neg_a=*/false, a, /*neg_b=*/false, b,
      /*c_mod=*/(short)0, c, /*reuse_a=*/false, /*reuse_b=*/false);
}

__device__ __forceinline__ float lrelu(float v) { return v > 0.f ? v : 0.2f * v; }

// ---------------------------------------------------------------------------
// Kernel 0: hp = proxies @ W (rows 0..99; rows 100..103 zero-padded),
//           sp[p,h] = <hp[p,h,:], att_src[h,:]>
// ---------------------------------------------------------------------------
__global__ __launch_bounds__(256)
void proxy_kernel(const float* __restrict__ proxies,
                  const float* __restrict__ W,
                  const float* __restrict__ att_src,
                  float* __restrict__ hp,
                  float* __restrict__ sp) {
  __shared__ float lds_row[HC_];
  const int b = blockIdx.x;
  const int t = threadIdx.x;
  if (b >= P_) {  // zero padding rows
    for (int i = t; i < HC_; i += 256) hp[b * HC_ + i] = 0.f;
    if (t < H_) sp[b * H_ + t] = 0.f;
    return;
  }
  for (int col = t; col < HC_; col += 256) {
    float acc = 0.f;
    for (int k = 0; k < E_; ++k) acc += proxies[b * E_ + k] * W[k * HC_ + col];
    hp[b * HC_ + col] = acc;
    lds_row[col] = acc;
  }
  __syncthreads();
  if (t < H_) {
    float s = 0.f;
    for (int c = 0; c < C_; ++c) s += lds_row[t * C_ + c] * att_src[t * C_ + c];
    sp[b * H_ + t] = s;
  }
}

// ---------------------------------------------------------------------------
// Kernel 1: fused per-node-tile pipeline (64 nodes / block, 8 waves).
//   h = x@W (WMMA, LDS) -> ss/ds -> softmax alpha (parallel, shfl reductions)
//   -> out = alpha@hp (WMMA) + self + bias + relu -> feats (LDS, coalesced
//   store) -> preds = feats@fc_W (WMMA, coalesced store via lds_x reuse).
// ---------------------------------------------------------------------------
__global__ __launch_bounds__(256)
void gat_node_kernel(const float* __restrict__ x,
                     const float* __restrict__ W,
                     const float* __restrict__ att_src,
                     const float* __restrict__ att_dst,
                     const float* __restrict__ bias,
                     const float* __restrict__ fc_W,
                     const float* __restrict__ fc_b,
                     const float* __restrict__ hp,
                     const float* __restrict__ sp,
                     float* __restrict__ preds_out,
                     float* __restrict__ feats_out) {
  __shared__ float lds_h[MT * HC_];       // 128 KB: h tile, later feats tile
  __shared__ float lds_x[MT * E_];        //  32 KB: x tile, later preds tile
  __shared__ float lds_alpha[MT * PPAD];  //  26 KB
  __shared__ float lds_ss[MT * H_];
  __shared__ float lds_ds[MT * H_];
  __shared__ float lds_asel[MT];

  const int tid   = threadIdx.x;
  const int wave  = tid >> 5;        // 0..7
  const int lane  = tid & 31;
  const int mrow  = lane & 15;       // A-row / B-col within a 16 tile
  const int khalf = lane >> 4;       // 0/1: K pair selector, D row-half
  const int tileBase = blockIdx.x * MT;

  // ---- stage x tile (64x128 f32, fully coalesced b128 loads) ----
  for (int i = tid; i < MT * E_ / 4; i += 256) {
    ((float4*)lds_x)[i] = ((const float4*)x)[(size_t)tileBase * (E_ / 4) + i];
  }
  __syncthreads();

  // ---- Step 1: h_tile = x_tile @ W  (M=64 -> 4 m-tiles, 32 n-tiles, K=128)
  {
    v8f acc[4][4];
#pragma unroll
    for (int m = 0; m < 4; ++m)
#pragma unroll
      for (int j = 0; j < 4; ++j) acc[m][j] = {};
    for (int kb = 0; kb < E_; kb += 4) {
      const int kk = kb + khalf * 2;
      v2f a[4];
#pragma unroll
      for (int m = 0; m < 4; ++m) {
        a[m].x = lds_x[(m * 16 + mrow) * E_ + kk];
        a[m].y = lds_x[(m * 16 + mrow) * E_ + kk + 1];
      }
#pragma unroll
      for (int j = 0; j < 4; ++j) {
        const int ncol = (wave * 4 + j) * 16 + mrow;
        v2f b;
        b.x = W[kk * HC_ + ncol];
        b.y = W[(kk + 1) * HC_ + ncol];
#pragma unroll
        for (int m = 0; m < 4; ++m) acc[m][j] = wmma_f32(a[m], b, acc[m][j]);
      }
    }
#pragma unroll
    for (int m = 0; m < 4; ++m)
#pragma unroll
      for (int j = 0; j < 4; ++j) {
        const int col = (wave * 4 + j) * 16 + mrow;
#pragma unroll
        for (int r = 0; r < 8; ++r) {
          const int row = m * 16 + r + khalf * 8;
          lds_h[row * HC_ + col] = acc[m][j][r];
        }
      }
  }
  __syncthreads();

  // ---- Step 2: ss/ds, one thread per (row, head) ----
  {
    const int row = tid >> 2, h = tid & 3;   // 64 rows x 4 heads = 256
    const float* hr = &lds_h[row * HC_ + h * C_];
    const float* as = &att_src[h * C_];
    const float* ad = &att_dst[h * C_];
    float s = 0.f, d = 0.f;
    for (int c = 0; c < C_; ++c) { const float v = hr[c]; s += v * as[c]; d += v * ad[c]; }
    lds_ss[row * H_ + h] = s;
    lds_ds[row * H_ + h] = d;
  }
  __syncthreads();

  // ---- Steps 3+4 per head ----
  for (int h = 0; h < H_; ++h) {
    // softmax over 101 logits, 4 lanes per row (26 entries each)
    {
      const int row = tid >> 2;          // 0..63
      const int sub = tid & 3;           // 0..3, within one wave
      const float dv = lds_ds[row * H_ + h];
      const float sv = lds_ss[row * H_ + h];
      float lg[26];
      float mx = -3.0e38f;
#pragma unroll
      for (int i = 0; i < 26; ++i) {
        const int idx = sub * 26 + i;
        float l = -3.0e38f;
        if (idx < P_)       l = lrelu(sp[idx * H_ + h] + dv);
        else if (idx == P_) l = lrelu(sv + dv);
        lg[i] = l;
        mx = fmaxf(mx, l);
      }
      mx = fmaxf(mx, __shfl_xor(mx, 1, 4));
      mx = fmaxf(mx, __shfl_xor(mx, 2, 4));
      float sum = 0.f;
      float es = 0.f;
#pragma unroll
      for (int i = 0; i < 26; ++i) {
        const int idx = sub * 26 + i;
        if (idx <= P_) {
          const float e = __expf(lg[i] - mx);
          sum += e;
          if (idx < P_) lds_alpha[row * PPAD + idx] = e;
          else          es = e;               // self
        } else if (idx < PPAD) {
          lds_alpha[row * PPAD + idx] = 0.f;  // padded K (hp rows are zero)
        }
      }
      sum += __shfl_xor(sum, 1, 4);
      sum += __shfl_xor(sum, 2, 4);
      const float inv = 1.f / sum;
#pragma unroll
      for (int i = 0; i < 26; ++i) {
        const int idx = sub * 26 + i;
        if (idx < P_) lds_alpha[row * PPAD + idx] *= inv;
      }
      if (sub == 3) lds_asel[row] = es * inv;  // lane owning idx==100
    }
    __syncthreads();

    // out_head(64x128): wave handles one 16-col n-tile, 4 m-tiles; K = 104
    v8f acc[4];
#pragma unroll
    for (int m = 0; m < 4; ++m) acc[m] = {};
    const int col = wave * 16 + mrow;
    for (int kb = 0; kb < PPAD; kb += 4) {
      const int kk = kb + khalf * 2;
      v2f b;
      b.x = hp[kk * HC_ + h * C_ + col];
      b.y = hp[(kk + 1) * HC_ + h * C_ + col];
#pragma unroll
      for (int m = 0; m < 4; ++m) {
        v2f a;
        a.x = lds_alpha[(m * 16 + mrow) * PPAD + kk];
        a.y = lds_alpha[(m * 16 + mrow) * PPAD + kk + 1];
        acc[m] = wmma_f32(a, b, acc[m]);
      }
    }
    // epilogue: + alpha_self * h_self + bias, ReLU; feats replace h in LDS
#pragma unroll
    for (int m = 0; m < 4; ++m) {
#pragma unroll
      for (int r = 0; r < 8; ++r) {
        const int row = m * 16 + r + khalf * 8;
        float v = acc[m][r] + lds_asel[row] * lds_h[row * HC_ + h * C_ + col];
        v += bias[h * C_ + col];
        v = fmaxf(v, 0.f);
        lds_h[row * HC_ + h * C_ + col] = v;
      }
    }
    __syncthreads();
  }

  // ---- coalesced feats store (64x512 contiguous floats) ----
  for (int i = tid; i < MT * HC_ / 4; i += 256) {
    ((float4*)(feats_out + (size_t)tileBase * HC_))[i] = ((const float4*)lds_h)[i];
  }

  // ---- Step 5: preds = feats @ fc_W + fc_b  (M=64, N=128, K=512) ----
  {
    v8f acc[4];
#pragma unroll
    for (int m = 0; m < 4; ++m) acc[m] = {};
    const int col = wave * 16 + mrow;
    for (int kb = 0; kb < HC_; kb += 4) {
      const int kk = kb + khalf * 2;
      v2f b;
      b.x = fc_W[kk * OUT_ + col];
      b.y = fc_W[(kk + 1) * OUT_ + col];
#pragma unroll
      for (int m = 0; m < 4; ++m) {
        v2f a;
        a.x = lds_h[(m * 16 + mrow) * HC_ + kk];
        a.y = lds_h[(m * 16 + mrow) * HC_ + kk + 1];
        acc[m] = wmma_f32(a, b, acc[m]);
      }
    }
    __syncthreads();  // lds_x dead; reuse as preds staging (64x128 f32)
#pragma unroll
    for (int m = 0; m < 4; ++m) {
#pragma unroll
      for (int r = 0; r < 8; ++r) {
        const int row = m * 16 + r + khalf * 8;
        lds_x[row * OUT_ + col] = acc[m][r] + fc_b[col];
      }
    }
  }
  __syncthreads();

  // ---- coalesced preds store (64x128 contiguous floats) ----
  for (int i = tid; i < MT * OUT_ / 4; i += 256) {
    ((float4*)(preds_out + (size_t)tileBase * OUT_))[i] = ((const float4*)lds_x)[i];
  }
}

// ---------------------------------------------------------------------------
extern "C" void kernel_launch(void* const* d_in, const int* in_sizes, int n_in,
                              void* d_out, int out_size, void* d_ws, size_t ws_size,
                              hipStream_t stream) {
  (void)in_sizes; (void)n_in; (void)out_size; (void)ws_size;
  const float* x       = (const float*)d_in[0];
  const float* proxies = (const float*)d_in[1];
  const float* W       = (const float*)d_in[2];
  const float* att_src = (const float*)d_in[3];
  const float* att_dst = (const float*)d_in[4];
  const float* bias    = (const float*)d_in[5];
  const float* fc_W    = (const float*)d_in[6];
  const float* fc_b    = (const float*)d_in[7];

  float* out = (float*)d_out;                    // [S*OUT preds][S*HC feats]
  float* hp  = (float*)d_ws;                     // PPAD * HC_ floats
  float* sp  = hp + PPAD * HC_;                  // PPAD * H_ floats

  proxy_kernel<<<PPAD, 256, 0, stream>>>(proxies, W, att_src, hp, sp);
  gat_node_kernel<<<S_ / MT, 256, 0, stream>>>(
      x, W, att_src, att_dst, bias, fc_W, fc_b, hp, sp,
      out, out + (size_t)S_ * OUT_);
}